// TNet_67250597921307
// MI455X (gfx1250) — compile-verified
//
#include <hip/hip_runtime.h>
#include <stdint.h>
#include <stddef.h>
#include <math.h>

#define XSTRIDE 1040   // padded row stride of activation matrix (>= 1028, mult of 16)
#define BATCHN  4096
#define NLAYERS 41

typedef __attribute__((ext_vector_type(2))) float v2f;
typedef __attribute__((ext_vector_type(8))) float v8f;

// ---------------------------------------------------------------------------
// Threefry-2x32 (20 rounds) — exact JAX PRNG
// ---------------------------------------------------------------------------
__host__ __device__ inline uint32_t rotl32(uint32_t v, int r) {
  return (v << r) | (v >> (32 - r));
}

#define TF_R4(ra, rb, rc, rd)                                                 \
  x0 += x1; x1 = rotl32(x1, ra); x1 ^= x0;                                    \
  x0 += x1; x1 = rotl32(x1, rb); x1 ^= x0;                                    \
  x0 += x1; x1 = rotl32(x1, rc); x1 ^= x0;                                    \
  x0 += x1; x1 = rotl32(x1, rd); x1 ^= x0;

__host__ __device__ inline void threefry2x32(uint32_t k0, uint32_t k1,
                                             uint32_t x0, uint32_t x1,
                                             uint32_t& o0, uint32_t& o1) {
  const uint32_t ks2 = k0 ^ k1 ^ 0x1BD11BDAu;
  x0 += k0; x1 += k1;
  TF_R4(13, 15, 26, 6)
  x0 += k1;  x1 += ks2 + 1u;
  TF_R4(17, 29, 16, 24)
  x0 += ks2; x1 += k0 + 2u;
  TF_R4(13, 15, 26, 6)
  x0 += k0;  x1 += k1 + 3u;
  TF_R4(17, 29, 16, 24)
  x0 += k1;  x1 += ks2 + 4u;
  TF_R4(13, 15, 26, 6)
  x0 += ks2; x1 += k0 + 5u;
  o0 = x0; o1 = x1;
}

// jax.random.uniform element at flat index f of a tensor with T total elements
// (T even for every draw in this problem). minval=1e-6, maxval=1.0.
__device__ inline float jax_uniform(uint32_t k0, uint32_t k1, uint32_t f, uint32_t T) {
  const uint32_t half = T >> 1;
  const uint32_t c0 = (f < half) ? f : (f - half);
  uint32_t o0, o1;
  threefry2x32(k0, k1, c0, c0 + half, o0, o1);
  const uint32_t bits = (f < half) ? o0 : o1;
  float u = __uint_as_float((bits >> 9) | 0x3f800000u) - 1.0f;
  u = u * (1.0f - 1e-6f) + 1e-6f;
  return fmaxf(u, 1e-6f);
}

__device__ inline float waveMax(float v) {
#pragma unroll
  for (int o = 16; o > 0; o >>= 1) v = fmaxf(v, __shfl_xor(v, o, 32));
  return v;
}
__device__ inline float waveSum(float v) {
#pragma unroll
  for (int o = 16; o > 0; o >>= 1) v += __shfl_xor(v, o, 32);
  return v;
}

// ---------------------------------------------------------------------------
// Per-layer weight preparation: Gumbel-softmax selections -> Wd, bias
// One wave32 per (slot, neuron) row.  Wd[(s*128+j)*XSTRIDE + k] = (p0-p1)*wlf
// bias[s*128+j] = sum_k p1*wlf
// ---------------------------------------------------------------------------
__global__ void __launch_bounds__(128) prep_kernel(
    const float* __restrict__ layer,   // (2, n, S, 2)
    const float* __restrict__ wlp,     // (2, n, L)
    float* __restrict__ Wd, float* __restrict__ bias,
    int n, int S, int L,
    uint32_t kin0, uint32_t kin1, uint32_t kdn0, uint32_t kdn1,
    uint32_t kup0, uint32_t kup1, uint32_t kwl0, uint32_t kwl1)
{
  __shared__ float wl_s[4][48];
  const int wave = threadIdx.x >> 5;
  const int lane = threadIdx.x & 31;
  const int row  = blockIdx.x * 4 + wave;
  if (row >= 2 * n) return;           // wave-uniform
  const int s = row / n;
  const int j = row - s * n;

  // ---- which_layers Gumbel-softmax (size L <= 41) ----
  {
    const uint32_t Twl = (uint32_t)(2 * n * L);
    float zz[2];
    float m = -INFINITY;
    int cw = 0;
    for (int t = lane; t < L; t += 32) {
      const float lg = fabsf(wlp[row * L + t]);
      const float u  = jax_uniform(kwl0, kwl1, (uint32_t)(row * L + t), Twl);
      const float z  = lg - logf(-logf(u));
      zz[cw++] = z;
      m = fmaxf(m, z);
    }
    m = waveMax(m);
    float sum = 0.f;
    for (int c = 0; c < cw; c++) sum += expf(zz[c] - m);
    sum = waveSum(sum);
    const float inv = 1.f / sum;
    int c = 0;
    for (int t = lane; t < L; t += 32) wl_s[wave][t] = expf(zz[c++] - m) * inv;
  }

  float biasAcc = 0.f;
  float* __restrict__ wrow = Wd + ((size_t)s * 128 + (size_t)j) * XSTRIDE;
  const float* __restrict__ lrow = layer + (size_t)row * (size_t)S * 2;

  auto doGroup = [&](int kstart, int kcnt, uint32_t K0, uint32_t K1,
                     uint32_t fbase, uint32_t T) {
    float z0[4], z1[4];                 // kcnt <= 128 -> <=4 pairs per lane
    float gm = -INFINITY;
    int c = 0;
    for (int kl = lane; kl < kcnt; kl += 32) {
      const int k = kstart + kl;
      const float u0 = jax_uniform(K0, K1, fbase + 2u * (uint32_t)kl,      T);
      const float u1 = jax_uniform(K0, K1, fbase + 2u * (uint32_t)kl + 1u, T);
      const float a = fabsf(lrow[2 * k])     - logf(-logf(u0));
      const float b = fabsf(lrow[2 * k + 1]) - logf(-logf(u1));
      z0[c] = a; z1[c] = b; c++;
      gm = fmaxf(gm, fmaxf(a, b));
    }
    gm = waveMax(gm);
    float sum = 0.f;
    for (int cc = 0; cc < c; cc++) sum += expf(z0[cc] - gm) + expf(z1[cc] - gm);
    sum = waveSum(sum);
    const float inv = 1.f / sum;
    c = 0;
    for (int kl = lane; kl < kcnt; kl += 32) {
      const int k = kstart + kl;
      const float p0 = expf(z0[c] - gm) * inv;
      const float p1 = expf(z1[c] - gm) * inv;
      c++;
      const int lo = (k < 128) ? 0
                   : ((k < 728) ? (1 + (k - 128) / 60) : (11 + (k - 728) / 10));
      const float wv = wl_s[wave][lo];
      wrow[k]  = (p0 - p1) * wv;
      biasAcc += p1 * wv;
    }
  };

  // input block: one softmax group over 128 k's (256 logits)
  doGroup(0, 128, kin0, kin1, (uint32_t)row * 256u, (uint32_t)(2 * n) * 256u);

  // down blocks: groups of 60 k's (120 logits)
  int dn_w = (S < 728 ? S : 728) - 128;
  const int Gdn = dn_w / 60;
  for (int g = 0; g < Gdn; g++)
    doGroup(128 + 60 * g, 60, kdn0, kdn1,
            ((uint32_t)row * (uint32_t)Gdn + (uint32_t)g) * 120u,
            (uint32_t)(2 * n * Gdn) * 120u);

  // up blocks: groups of 10 k's (20 logits)
  int up_w = S - 728; if (up_w < 0) up_w = 0;
  const int Gup = up_w / 10;
  for (int g = 0; g < Gup; g++)
    doGroup(728 + 10 * g, 10, kup0, kup1,
            ((uint32_t)row * (uint32_t)Gup + (uint32_t)g) * 20u,
            (uint32_t)(2 * n * Gup) * 20u);

  biasAcc = waveSum(biasAcc);
  if (lane == 0) bias[s * 128 + j] = biasAcc;
}

// ---------------------------------------------------------------------------
// GEMM via V_WMMA_F32_16X16X4_F32 (exact fp32):
// O[slot][m][j] = bias[slot][j] + sum_k X[m][k] * Wd[slot][j][k]
// One wave per 16x16 tile; A = X rows (M x K), B = Wd rows (N x K, i.e. B^T).
// ---------------------------------------------------------------------------
__global__ void __launch_bounds__(256) gemm_kernel(
    const float* __restrict__ X, const float* __restrict__ Wd,
    const float* __restrict__ bias, float* __restrict__ O,
    int Kpad, int ntiles, int nslots)
{
  const int wid   = (int)((blockIdx.x * blockDim.x + threadIdx.x) >> 5);
  const int lane  = (int)(threadIdx.x & 31);
  const int mtile = wid & 255;
  const int rest  = wid >> 8;
  const int ntile = rest % ntiles;
  const int slot  = rest / ntiles;
  if (slot >= nslots) return;         // grid sized exactly; never taken

  const int m0   = mtile << 4;
  const int n0   = ntile << 4;
  const int half = lane >> 4;         // 0: K+0/K+1, 1: K+2/K+3
  const int lm   = lane & 15;

  const float* __restrict__ xrow =
      X + (size_t)(m0 + lm) * XSTRIDE + 2 * half;
  const float* __restrict__ wrow =
      Wd + ((size_t)slot * 128 + (size_t)(n0 + lm)) * XSTRIDE + 2 * half;

  v8f acc0 = {};
  v8f acc1 = {};
  int kk = 0;
  for (; kk + 8 <= Kpad; kk += 8) {
    v2f a0 = *(const v2f*)(xrow + kk);
    v2f b0 = *(const v2f*)(wrow + kk);
    v2f a1 = *(const v2f*)(xrow + kk + 4);
    v2f b1 = *(const v2f*)(wrow + kk + 4);
    acc0 = __builtin_amdgcn_wmma_f32_16x16x4_f32(false, a0, false, b0,
                                                 (short)0, acc0, false, false);
    acc1 = __builtin_amdgcn_wmma_f32_16x16x4_f32(false, a1, false, b1,
                                                 (short)0, acc1, false, false);
  }
  if (kk < Kpad) {
    v2f a0 = *(const v2f*)(xrow + kk);
    v2f b0 = *(const v2f*)(wrow + kk);
    acc0 = __builtin_amdgcn_wmma_f32_16x16x4_f32(false, a0, false, b0,
                                                 (short)0, acc0, false, false);
  }

  const float bv = bias[slot * 128 + n0 + lm];
  float* __restrict__ orow = O + (size_t)slot * ((size_t)BATCHN * 128);
#pragma unroll
  for (int r = 0; r < 8; r++) {
    const int m = m0 + r + 8 * half;  // C/D layout: VGPR r -> M=r (lanes 0-15), M=r+8 (16-31)
    orow[(size_t)m * 128 + (n0 + lm)] = acc0[r] + acc1[r] + bv;
  }
}

// ---------------------------------------------------------------------------
__global__ void initX_kernel(const float* __restrict__ x, float* __restrict__ X) {
  const int idx = blockIdx.x * blockDim.x + threadIdx.x;   // BATCHN*128 threads
  const int r = idx >> 7, c = idx & 127;
  X[(size_t)r * XSTRIDE + c] = x[idx];
}

__global__ void combine_kernel(const float* __restrict__ O, float* __restrict__ X,
                               int S, int n, int total) {
  const int idx = blockIdx.x * blockDim.x + threadIdx.x;
  if (idx >= total) return;
  const int r = idx / n, j = idx - r * n;
  const float a = O[(size_t)r * 128 + j];
  const float b = O[(size_t)BATCHN * 128 + (size_t)r * 128 + j];
  X[(size_t)r * XSTRIDE + S + j] = 1.0f - a * b;
}

// ---------------------------------------------------------------------------
extern "C" void kernel_launch(void* const* d_in, const int* in_sizes, int n_in,
                              void* d_out, int out_size, void* d_ws, size_t ws_size,
                              hipStream_t stream) {
  (void)in_sizes; (void)n_in; (void)out_size; (void)ws_size;

  int LL[NLAYERS + 1];
  LL[0] = 128;
  for (int t = 1; t <= 10; t++) LL[t] = 60;
  for (int t = 11; t <= 40; t++) LL[t] = 10;
  LL[NLAYERS] = 128;

  const float* x = (const float*)d_in[0];

  float* X    = (float*)d_ws;                          // BATCHN * XSTRIDE
  float* Wd   = X + (size_t)BATCHN * XSTRIDE;          // 2 * 128 * XSTRIDE
  float* bias = Wd + (size_t)2 * 128 * XSTRIDE;        // 256
  float* O    = bias + 256;                            // 2 * BATCHN * 128

  // Reset all state every call (graph replays reuse buffers).
  hipMemsetAsync(X,  0, sizeof(float) * (size_t)BATCHN * XSTRIDE, stream);
  hipMemsetAsync(Wd, 0, sizeof(float) * (size_t)2 * 128 * XSTRIDE, stream);
  hipMemsetAsync(bias, 0, sizeof(float) * 256, stream);
  initX_kernel<<<(BATCHN * 128) / 256, 256, 0, stream>>>(x, X);

  const uint32_t base0 = 0u, base1 = 42u;   // jax.random.key(42)
  int S = 128;
  for (int i = 0; i < NLAYERS; i++) {
    const int n = LL[i + 1];

    // ki = fold_in(base, i);  k_in,k_dn,k_up,k_wl = split(ki, 4)
    uint32_t ki0, ki1;
    threefry2x32(base0, base1, 0u, (uint32_t)i, ki0, ki1);
    uint32_t a0, a1, b0, b1, c0, c1, d0, d1;
    threefry2x32(ki0, ki1, 0u, 4u, a0, a1);
    threefry2x32(ki0, ki1, 1u, 5u, b0, b1);
    threefry2x32(ki0, ki1, 2u, 6u, c0, c1);
    threefry2x32(ki0, ki1, 3u, 7u, d0, d1);
    // split() reshape interleaving: key_j = (out[2j], out[2j+1]) of concat(o0s, o1s)
    const uint32_t kin0 = a0, kin1 = b0;
    const uint32_t kdn0 = c0, kdn1 = d0;
    const uint32_t kup0 = a1, kup1 = b1;
    const uint32_t kwl0 = c1, kwl1 = d1;

    const float* layer = (const float*)d_in[1 + i];
    const float* wlp   = (const float*)d_in[1 + NLAYERS + i];

    const int rows = 2 * n;
    prep_kernel<<<(rows + 3) / 4, 128, 0, stream>>>(
        layer, wlp, Wd, bias, n, S, i + 1,
        kin0, kin1, kdn0, kdn1, kup0, kup1, kwl0, kwl1);

    const int  Npad   = (n + 15) & ~15;
    const int  ntiles = Npad >> 4;
    const int  Kpad   = (S + 3) & ~3;
    const bool last   = (i == NLAYERS - 1);
    const int  nslots = last ? 1 : 2;
    float*     Optr   = last ? (float*)d_out : O;
    const int  waves  = 256 * ntiles * nslots;
    gemm_kernel<<<waves / 8, 256, 0, stream>>>(X, Wd, bias, Optr, Kpad, ntiles, nslots);

    if (!last) {
      const int total = BATCHN * n;
      combine_kernel<<<(total + 255) / 256, 256, 0, stream>>>(O, X, S, n, total);
      S += n;
    }
  }
}